// GCN_my_78365973283345
// MI455X (gfx1250) — compile-verified
//
#include <hip/hip_runtime.h>

// GCN forward for MI455X (gfx1250).
// DIMS fixed by the reference: 96 -> 96 -> 48 -> 1, N=100000 (divisible by 16), E=800000.

#define DIM0 96
#define DIM1 96
#define DIM2 48
#define BN_EPS 1e-5f

typedef __attribute__((ext_vector_type(2))) float v2f;
typedef __attribute__((ext_vector_type(8))) float v8f;

__device__ __forceinline__ float sigmoidf_(float x) {
    return 1.0f / (1.0f + __expf(-x));
}

// ---------------- utility kernels ----------------

__global__ void k_zero(float* __restrict__ p, long n) {
    long i = (long)blockIdx.x * blockDim.x + threadIdx.x;
    long stride = (long)gridDim.x * blockDim.x;
    for (; i < n; i += stride) p[i] = 0.0f;
}

__global__ void k_sigmoid_arr(const float* __restrict__ in, float* __restrict__ out, long n) {
    long i = (long)blockIdx.x * blockDim.x + threadIdx.x;
    long stride = (long)gridDim.x * blockDim.x;
    for (; i < n; i += stride) out[i] = sigmoidf_(in[i]);
}

// degrees via hardware fp32 atomics (GLOBAL_ATOMIC_ADD_F32; L2-resident RMW)
__global__ void k_degrees(const int* __restrict__ src, const int* __restrict__ dst,
                          float* __restrict__ deg_out, float* __restrict__ deg_in, int E) {
    int i = blockIdx.x * blockDim.x + threadIdx.x;
    int stride = gridDim.x * blockDim.x;
    for (; i < E; i += stride) {
        unsafeAtomicAdd(&deg_out[src[i]], 1.0f);
        unsafeAtomicAdd(&deg_in[dst[i]], 1.0f);
    }
}

__global__ void k_rsqrt_deg(float* __restrict__ p, long n) {
    long i = (long)blockIdx.x * blockDim.x + threadIdx.x;
    long stride = (long)gridDim.x * blockDim.x;
    for (; i < n; i += stride) p[i] = rsqrtf(fmaxf(p[i], 1.0f));
}

// agg[dst] += x[src] * norm_src[src] ; one thread per (edge, float4-chunk),
// consecutive threads share an edge -> coalesced 384B gather per edge.
__global__ void k_aggregate(const float* __restrict__ x, const int* __restrict__ src,
                            const int* __restrict__ dst, const float* __restrict__ nsrc,
                            float* __restrict__ agg, int E, int D4, int D) {
    int total = E * D4;
    int i = blockIdx.x * blockDim.x + threadIdx.x;
    int stride = gridDim.x * blockDim.x;
    for (; i < total; i += stride) {
        int e = i / D4;
        int c = i - e * D4;
        int s = src[e];
        int d = dst[e];
        float ns = nsrc[s];
        float4 v = ((const float4*)(x + (long)s * D))[c];
        float* ap = agg + (long)d * D + 4 * c;
        unsafeAtomicAdd(ap + 0, v.x * ns);
        unsafeAtomicAdd(ap + 1, v.y * ns);
        unsafeAtomicAdd(ap + 2, v.z * ns);
        unsafeAtomicAdd(ap + 3, v.w * ns);
    }
}

// ---------------- f32 WMMA GEMM: out = (A * nd[:,None]) @ W + bias ----------------
// V_WMMA_F32_16X16X4_F32, one wave per 16x16 output tile, K stepped by 4.
// A-layout (16x4): lanes 0-15 K={kk,kk+1}; lanes 16-31 K={kk+2,kk+3}; M = lane&15.
// B-layout (4x16): mirrored. C/D: VGPR r -> M=r (lanes 0-15) / M=r+8 (lanes 16-31), N=lane&15.
__global__ void k_gemm_wmma(const float* __restrict__ A, const float* __restrict__ nd,
                            const float* __restrict__ W, const float* __restrict__ bias,
                            float* __restrict__ out, int nrows, int K, int ncols) {
    int lane = threadIdx.x & 31;
    int wave = threadIdx.x >> 5;
    int half = lane >> 4;          // 0: K pair {0,1}, 1: K pair {2,3}
    int lm   = lane & 15;
    int mtiles = nrows >> 4;
    int ntiles = ncols >> 4;
    int tile = blockIdx.x * (blockDim.x >> 5) + wave;
    if (tile >= mtiles * ntiles) return;   // wave-uniform: EXEC stays all-1s for WMMA
    int mt = tile / ntiles;
    int nt = tile - mt * ntiles;
    int row  = (mt << 4) + lm;
    int col0 = nt << 4;

    const float* arow = A + (long)row * K + 2 * half;
    float scale = nd[row];
    float bv = bias[col0 + lm];

    v8f c = {0.f, 0.f, 0.f, 0.f, 0.f, 0.f, 0.f, 0.f};
    for (int kk = 0; kk < K; kk += 4) {
        float2 av = *(const float2*)(arow + kk);   // 8B aligned (K even, kk even)
        v2f a;
        a.x = av.x * scale;
        a.y = av.y * scale;
        v2f b;
        b.x = W[(kk + 2 * half + 0) * ncols + col0 + lm];
        b.y = W[(kk + 2 * half + 1) * ncols + col0 + lm];
        c = __builtin_amdgcn_wmma_f32_16x16x4_f32(false, a, false, b, (short)0, c,
                                                  false, false);
    }
#pragma unroll
    for (int r = 0; r < 8; ++r) {
        int m = (mt << 4) + r + 8 * half;
        out[(long)m * ncols + col0 + lm] = c[r] + bv;
    }
}

// ---------------- layer-2 GEMV (out_d == 1): wave per row ----------------
__global__ void k_gemv(const float* __restrict__ A, const float* __restrict__ nd,
                       const float* __restrict__ W, const float* __restrict__ bias,
                       float* __restrict__ out, int n, int K) {
    int gid = blockIdx.x * blockDim.x + threadIdx.x;
    int row = gid >> 5;
    int lane = gid & 31;
    if (row >= n) return;
    float s = 0.0f;
    for (int k = lane; k < K; k += 32) s += A[(long)row * K + k] * W[k];
    for (int off = 16; off > 0; off >>= 1) s += __shfl_down(s, off, 32);
    if (lane == 0) out[row] = s * nd[row] + bias[0];
}

// ---------------- batch-norm stats: one block per column ----------------
// Also emits hid = sigmoid(pre-BN outs[row 0]) into the two duplicated output slots.
__global__ void k_bn_stats(const float* __restrict__ outs, int n, int ncols,
                           float* __restrict__ mean, float* __restrict__ rstd,
                           float* __restrict__ hidA, float* __restrict__ hidB) {
    int col = blockIdx.x;
    __shared__ double ssum[256];
    __shared__ double ssq[256];
    double s = 0.0, q = 0.0;
    for (int i = threadIdx.x; i < n; i += blockDim.x) {
        float v = outs[(long)i * ncols + col];
        s += (double)v;
        q += (double)v * (double)v;
    }
    ssum[threadIdx.x] = s;
    ssq[threadIdx.x] = q;
    __syncthreads();
    for (int off = blockDim.x >> 1; off > 0; off >>= 1) {
        if ((int)threadIdx.x < off) {
            ssum[threadIdx.x] += ssum[threadIdx.x + off];
            ssq[threadIdx.x]  += ssq[threadIdx.x + off];
        }
        __syncthreads();
    }
    if (threadIdx.x == 0) {
        double m = ssum[0] / n;
        double var = ssq[0] / n - m * m;
        mean[col] = (float)m;
        rstd[col] = rsqrtf((float)var + BN_EPS);
        float h = sigmoidf_(outs[col]);  // row SPE_IDX = 0
        hidA[col] = h;
        hidB[col] = h;
    }
}

// y = (x - mean)*rstd*gamma + beta, then ReLU (act=0) or sigmoid (act=1)
__global__ void k_bn_apply(const float* __restrict__ outs, const float* __restrict__ mean,
                           const float* __restrict__ rstd, const float* __restrict__ gamma,
                           const float* __restrict__ beta, float* __restrict__ xn,
                           long total, int ncols, int act) {
    long i = (long)blockIdx.x * blockDim.x + threadIdx.x;
    long stride = (long)gridDim.x * blockDim.x;
    for (; i < total; i += stride) {
        int col = (int)(i % ncols);
        float y = (outs[i] - mean[col]) * rstd[col] * gamma[col] + beta[col];
        xn[i] = act ? sigmoidf_(y) : fmaxf(y, 0.0f);
    }
}

// ---------------- host launcher ----------------

static inline int nblk(long n, int t) { return (int)((n + t - 1) / t); }

extern "C" void kernel_launch(void* const* d_in, const int* in_sizes, int n_in,
                              void* d_out, int out_size, void* d_ws, size_t ws_size,
                              hipStream_t stream) {
    const float* x0  = (const float*)d_in[0];
    const int*   src = (const int*)d_in[1];
    const int*   dst = (const int*)d_in[2];
    const float* W0  = (const float*)d_in[3];
    const float* b0  = (const float*)d_in[4];
    const float* g0  = (const float*)d_in[5];
    const float* bt0 = (const float*)d_in[6];
    const float* W1  = (const float*)d_in[7];
    const float* b1  = (const float*)d_in[8];
    const float* g1  = (const float*)d_in[9];
    const float* bt1 = (const float*)d_in[10];
    const float* W2  = (const float*)d_in[11];
    const float* b2  = (const float*)d_in[12];
    const float* g2  = (const float*)d_in[13];
    const float* bt2 = (const float*)d_in[14];

    const int N = in_sizes[0] / DIM0;   // 100000 (multiple of 16)
    const int E = in_sizes[1];          // 800000
    const long ND0 = (long)N * DIM0;

    // workspace layout (floats)
    float* ws = (float*)d_ws;
    float* bufA = ws;                 // agg buffer   [N,96]
    float* bufB = bufA + ND0;         // GEMM output  [N,96]
    float* bufC = bufB + ND0;         // activated x  [N,96]
    float* nsrc = bufC + ND0;         // norm_src [N]
    float* ndst = nsrc + N;           // norm_dst [N] (contiguous with nsrc)
    float* mean = ndst + N;           // [96]
    float* rstd = mean + DIM0;        // [96]

    // output layout (floats): outs[N] | sigmoid(inputs)[N*96] | hid0 x2 | hid1 x2 | hid2 x2
    float* out       = (float*)d_out;
    float* out_final = out;
    float* out_sig   = out + N;
    float* hid0a = out + N + ND0;
    float* hid0b = hid0a + DIM1;
    float* hid1a = hid0b + DIM1;
    float* hid1b = hid1a + DIM2;
    float* hid2a = hid1b + DIM2;
    float* hid2b = hid2a + 1;

    const int T = 256;

    // degrees -> rsqrt norms (nsrc/ndst contiguous: single zero + single rsqrt pass)
    k_zero<<<nblk(2L * N, T), T, 0, stream>>>(nsrc, 2L * N);
    k_degrees<<<nblk(E, T), T, 0, stream>>>(src, dst, nsrc, ndst, E);
    k_rsqrt_deg<<<nblk(2L * N, T), T, 0, stream>>>(nsrc, 2L * N);

    // hids[0] = sigmoid(inputs)
    k_sigmoid_arr<<<nblk(ND0, T), T, 0, stream>>>(x0, out_sig, ND0);

    // ---- layer 0: 96 -> 96 ----
    k_zero<<<nblk(ND0, T), T, 0, stream>>>(bufA, ND0);
    k_aggregate<<<nblk((long)E * (DIM0 / 4), T), T, 0, stream>>>(x0, src, dst, nsrc, bufA,
                                                                 E, DIM0 / 4, DIM0);
    {
        long tiles = (long)(N / 16) * (DIM1 / 16);
        k_gemm_wmma<<<nblk(tiles * 32, T), T, 0, stream>>>(bufA, ndst, W0, b0, bufB,
                                                           N, DIM0, DIM1);
    }
    k_bn_stats<<<DIM1, 256, 0, stream>>>(bufB, N, DIM1, mean, rstd, hid0a, hid0b);
    k_bn_apply<<<nblk((long)N * DIM1, T), T, 0, stream>>>(bufB, mean, rstd, g0, bt0, bufC,
                                                          (long)N * DIM1, DIM1, /*relu*/0);

    // ---- layer 1: 96 -> 48 ----
    k_zero<<<nblk(ND0, T), T, 0, stream>>>(bufA, ND0);
    k_aggregate<<<nblk((long)E * (DIM1 / 4), T), T, 0, stream>>>(bufC, src, dst, nsrc, bufA,
                                                                 E, DIM1 / 4, DIM1);
    {
        long tiles = (long)(N / 16) * (DIM2 / 16);
        k_gemm_wmma<<<nblk(tiles * 32, T), T, 0, stream>>>(bufA, ndst, W1, b1, bufB,
                                                           N, DIM1, DIM2);
    }
    k_bn_stats<<<DIM2, 256, 0, stream>>>(bufB, N, DIM2, mean, rstd, hid1a, hid1b);
    k_bn_apply<<<nblk((long)N * DIM2, T), T, 0, stream>>>(bufB, mean, rstd, g1, bt1, bufC,
                                                          (long)N * DIM2, DIM2, /*relu*/0);

    // ---- layer 2: 48 -> 1 (GEMV) ----
    k_zero<<<nblk((long)N * DIM2, T), T, 0, stream>>>(bufA, (long)N * DIM2);
    k_aggregate<<<nblk((long)E * (DIM2 / 4), T), T, 0, stream>>>(bufC, src, dst, nsrc, bufA,
                                                                 E, DIM2 / 4, DIM2);
    k_gemv<<<nblk((long)N * 32, T), T, 0, stream>>>(bufA, ndst, W2, b2, bufB, N, DIM2);
    k_bn_stats<<<1, 256, 0, stream>>>(bufB, N, 1, mean, rstd, hid2a, hid2b);
    k_bn_apply<<<nblk((long)N, T), T, 0, stream>>>(bufB, mean, rstd, g2, bt2, out_final,
                                                   (long)N, 1, /*sigmoid*/1);
}